// NumericalRiskBudgeting_43422119362991
// MI455X (gfx1250) — compile-verified
//
#include <hip/hip_runtime.h>
#include <hip/hip_bf16.h>
#include <math.h>

// CDNA5 / gfx1250: wave32, WMMA f32 16x16x4, Sigma register-resident.
typedef float v2f __attribute__((ext_vector_type(2)));
typedef float v8f __attribute__((ext_vector_type(8)));

#define N_ASSETS   64
#define SROW       68      // padded LDS row stride (dwords): bank-conflict-free strided reads
#define N_STEPS    100
#define BATCH_DIV  (64 * 64)

__device__ __forceinline__ float wave_sum32(float x) {
    // wave32 butterfly allreduce
    #pragma unroll
    for (int off = 16; off > 0; off >>= 1)
        x += __shfl_xor(x, off, 32);
    return x;
}

__launch_bounds__(32, 1)
__global__ void risk_budget_kernel(const float* __restrict__ sigma,
                                   const float* __restrict__ bvec,
                                   float* __restrict__ out,
                                   float inv_batch)
{
    __shared__ float lds_sig[N_ASSETS * SROW]; // Sigma staging (dead after reg preload)
    __shared__ float lds_x[N_ASSETS];          // broadcast vector for matvec B operand
    __shared__ float lds_y[N_ASSETS];          // matvec result exchange

    const int lane  = threadIdx.x;     // 0..31, single wave per block
    const int batch = blockIdx.x;
    const int lm    = lane & 15;       // row within A tile
    const int h     = lane >> 4;       // half select (K pair / row-half)

    // ---- stage Sigma: coalesced HBM -> LDS (Sigma read from HBM exactly once) ----
    const float* S = sigma + (size_t)batch * (N_ASSETS * N_ASSETS);
    #pragma unroll
    for (int it = 0; it < 32; ++it) {
        int idx = it * 128 + lane * 4;          // float4 per lane per iter
        int row = idx >> 6;
        int col = idx & 63;
        float4 v = *(const float4*)(S + idx);
        lds_sig[row * SROW + col + 0] = v.x;
        lds_sig[row * SROW + col + 1] = v.y;
        lds_sig[row * SROW + col + 2] = v.z;
        lds_sig[row * SROW + col + 3] = v.w;
    }
    __syncthreads();

    // ---- preload Sigma into WMMA A-operand register tiles (128 VGPRs) ----
    // A (16x4 f32): lane lm holds row (16r+lm); VGPR0 = K(2h), VGPR1 = K(2h+1), K base = 4k.
    v2f a[4][16];
    #pragma unroll
    for (int r = 0; r < 4; ++r) {
        #pragma unroll
        for (int k = 0; k < 16; ++k) {
            int base = (16 * r + lm) * SROW + 4 * k + 2 * h;   // 8B-aligned -> ds_load_b64
            v2f t;
            t.x = lds_sig[base];
            t.y = lds_sig[base + 1];
            a[r][k] = t;
        }
    }
    __syncthreads();

    // y = Sigma * x on the matrix pipe. B (4x16) columns all = x -> C columns all = y.
    // B layout: VGPR0 rows {0 (lanes0-15), 2 (lanes16-31)}, VGPR1 rows {1,3}:
    // lane's float2 = x[4k + 2h .. 4k + 2h + 1] (contiguous LDS read).
    auto matvec = [&](float x0, float x1, float& y0, float& y1) {
        lds_x[lane]      = x0;
        lds_x[lane + 32] = x1;
        __syncthreads();
        v2f bop[16];
        #pragma unroll
        for (int k = 0; k < 16; ++k) {
            int o = 4 * k + 2 * h;
            v2f t;
            t.x = lds_x[o];
            t.y = lds_x[o + 1];
            bop[k] = t;
        }
        // 4 independent accumulator chains, interleaved (k outer, r inner):
        // consecutive v_wmma ops are never RAW-dependent -> matrix pipe stays full.
        v8f c[4];
        #pragma unroll
        for (int r = 0; r < 4; ++r)
            c[r] = (v8f){0.f, 0.f, 0.f, 0.f, 0.f, 0.f, 0.f, 0.f};
        #pragma unroll
        for (int k = 0; k < 16; ++k) {
            #pragma unroll
            for (int r = 0; r < 4; ++r) {
                c[r] = __builtin_amdgcn_wmma_f32_16x16x4_f32(
                           false, a[r][k], false, bop[k], (short)0, c[r], false, false);
            }
        }
        // C layout: VGPR i = row (16r + i + 8h); all columns identical, so every lane
        // in a half holds the value -> unconditional stores (no EXEC branching);
        // redundant same-address same-value writes are benign; pairs merge to b64.
        #pragma unroll
        for (int r = 0; r < 4; ++r) {
            int base = 16 * r + 8 * h;
            #pragma unroll
            for (int i = 0; i < 8; ++i)
                lds_y[base + i] = c[r][i];
        }
        __syncthreads();
        y0 = lds_y[lane];
        y1 = lds_y[lane + 32];
    };

    // ---- per-lane optimizer state: elements (lane) and (lane+32) ----
    const float b0 = bvec[(size_t)batch * N_ASSETS + lane];
    const float b1 = bvec[(size_t)batch * N_ASSETS + lane + 32];
    float w0 = 1.0f / 64.0f, w1 = 1.0f / 64.0f;
    float m0 = 0.f, m1 = 0.f, v0 = 0.f, v1 = 0.f;
    float p1 = 1.0f, p2 = 1.0f;               // running B1^t, B2^t

    const float PEN  = 100.0f;
    const float LR   = 0.01f;
    const float B1c  = 0.9f, B2c = 0.999f;
    const float EPSc = 1e-8f;

    for (int t = 0; t < N_STEPS; ++t) {
        // ---- gradient of the loss at current w ----
        float u0, u1;
        matvec(w0, w1, u0, u1);                       // u = Sigma w
        float pr  = wave_sum32(w0 * u0 + w1 * u1);    // w^T Sigma w
        float mr0 = 2.0f * u0, mr1 = 2.0f * u1;       // marginal risk
        float e0  = w0 * mr0 - b0 * pr;               // rc - b*pr
        float e1  = w1 * mr1 - b1 * pr;
        float eb  = wave_sum32(e0 * b0 + e1 * b1);    // e . b
        float sumw = wave_sum32(w0 + w1);             // for |sum(w)-1| subgradient
        float s0, s1;
        matvec(e0 * w0, e1 * w1, s0, s1);             // s = Sigma (e .* w)

        float sgn = (sumw >= 1.0f) ? 1.0f : -1.0f;
        float g0 = inv_batch * (2.0f * e0 * mr0 + 4.0f * s0 - 2.0f * mr0 * eb)
                 + PEN * inv_batch * sgn
                 + (PEN * inv_batch / 64.0f) * ((w0 > 1.0f ? 1.0f : 0.0f) - (w0 < 0.0f ? 1.0f : 0.0f));
        float g1 = inv_batch * (2.0f * e1 * mr1 + 4.0f * s1 - 2.0f * mr1 * eb)
                 + PEN * inv_batch * sgn
                 + (PEN * inv_batch / 64.0f) * ((w1 > 1.0f ? 1.0f : 0.0f) - (w1 < 0.0f ? 1.0f : 0.0f));

        // ---- Adam update with bias correction (shared denominators computed once) ----
        p1 *= B1c;  p2 *= B2c;
        m0 = B1c * m0 + (1.0f - B1c) * g0;
        m1 = B1c * m1 + (1.0f - B1c) * g1;
        v0 = B2c * v0 + (1.0f - B2c) * g0 * g0;
        v1 = B2c * v1 + (1.0f - B2c) * g1 * g1;
        float bc1 = 1.0f / (1.0f - p1);
        float bc2 = 1.0f / (1.0f - p2);
        float mh0 = m0 * bc1, mh1 = m1 * bc1;
        float vh0 = v0 * bc2, vh1 = v1 * bc2;
        w0 -= LR * mh0 / (sqrtf(vh0) + EPSc);
        w1 -= LR * mh1 / (sqrtf(vh1) + EPSc);

        // ---- projection: clip to [0,1], renormalize rows to sum 1 ----
        w0 = fminf(fmaxf(w0, 0.0f), 1.0f);
        w1 = fminf(fmaxf(w1, 0.0f), 1.0f);
        float rs = 1.0f / wave_sum32(w0 + w1);
        w0 *= rs;  w1 *= rs;
    }

    out[(size_t)batch * N_ASSETS + lane]      = w0;
    out[(size_t)batch * N_ASSETS + lane + 32] = w1;
}

extern "C" void kernel_launch(void* const* d_in, const int* in_sizes, int n_in,
                              void* d_out, int out_size, void* d_ws, size_t ws_size,
                              hipStream_t stream) {
    const float* sigma = (const float*)d_in[0];   // [B, 64, 64] f32
    const float* b     = (const float*)d_in[1];   // [B, 64] f32
    float* out         = (float*)d_out;           // [B, 64] f32
    int batch = in_sizes[0] / BATCH_DIV;
    float inv_batch = 1.0f / (float)batch;
    // one wave32 per batch element; Sigma resident in 128 VGPRs for all 100 steps
    risk_budget_kernel<<<batch, 32, 0, stream>>>(sigma, b, out, inv_batch);
}